// CumsumInterpolationEmbedding_46935402611135
// MI455X (gfx1250) — compile-verified
//
#include <hip/hip_runtime.h>

typedef __attribute__((ext_vector_type(2))) float v2f;
typedef __attribute__((ext_vector_type(4))) float v4f;
typedef __attribute__((ext_vector_type(8))) float v8f;

#define NUM_EMB 1024
#define EMB_DIM 128
#define LN_EPS  1e-5f

// ---------------------------------------------------------------------------
// Kernel 1: inclusive prefix-sum over rows of E[1024][128] expressed as
// blocked tril(ones(16)) @ E via V_WMMA_F32_16X16X4_F32 (full f32 precision).
// One workgroup, 8 waves; wave w owns columns [16w, 16w+16) and scans the 64
// row-blocks sequentially, keeping the per-column carry in registers.
// A-matrix 16x4 layout: lanes0-15 M=lane, lanes16-31 M=lane-16;
//   VGPR0 = K (+2 for upper half), VGPR1 = K+1. B 4x16 mirrors this.
// C/D 16x16 layout: VGPR v = row v (lanes0-15) / row v+8 (lanes16-31).
// ---------------------------------------------------------------------------
__global__ __launch_bounds__(256) void cie_cumsum_wmma_kernel(
    const float* __restrict__ E, float* __restrict__ raw) {
  const int tid  = threadIdx.x;
  const int wave = tid >> 5;
  const int lane = tid & 31;
  const int m    = lane & 15;   // M for A; N (column) for B/C/D
  const int h    = lane >> 4;   // half-wave select
  const int col  = wave * 16 + m;

  float carry = 0.0f;
  for (int b = 0; b < 64; ++b) {
    const int rowbase = b * 16;
    v8f acc = {carry, carry, carry, carry, carry, carry, carry, carry};
#pragma unroll
    for (int k0 = 0; k0 < 4; ++k0) {
      const int kA = k0 * 4 + 2 * h;
      v2f a, bb;
      a.x  = (kA     <= m) ? 1.0f : 0.0f;   // tril(ones(16)) chunk
      a.y  = (kA + 1 <= m) ? 1.0f : 0.0f;
      bb.x = E[(rowbase + kA    ) * EMB_DIM + col];
      bb.y = E[(rowbase + kA + 1) * EMB_DIM + col];
      acc  = __builtin_amdgcn_wmma_f32_16x16x4_f32(
                 false, a, false, bb, (short)0, acc, false, false);
    }
#pragma unroll
    for (int v = 0; v < 8; ++v) {
      raw[(rowbase + h * 8 + v) * EMB_DIM + col] = acc[v];
    }
    // Row 15 of this block = VGPR7 in upper-half lanes -> next block's carry.
    carry = __shfl(acc[7], 16 + m, 32);
  }
}

// ---------------------------------------------------------------------------
// Kernel 2: per-row LayerNorm (dim 128) + interpolation term, in place.
// One wave32 per row; float4 per lane; wave shuffle reductions.
// ---------------------------------------------------------------------------
__device__ __forceinline__ float wave_sum32(float v) {
#pragma unroll
  for (int off = 16; off > 0; off >>= 1) v += __shfl_xor(v, off, 32);
  return v;
}

__global__ __launch_bounds__(256) void cie_ln_interp_kernel(
    float* __restrict__ tbl, const float* __restrict__ IE,
    const float* __restrict__ gamma, const float* __restrict__ beta) {
  const int wave = threadIdx.x >> 5;
  const int lane = threadIdx.x & 31;
  const int row  = blockIdx.x * 8 + wave;
  if (row >= NUM_EMB) return;

  const v4f* g4 = (const v4f*)gamma;
  const v4f* b4 = (const v4f*)beta;
  const v4f* i0 = (const v4f*)IE;
  const v4f* i1 = (const v4f*)(IE + EMB_DIM);
  v4f* t4 = (v4f*)(tbl + (size_t)row * EMB_DIM);

  v4f x = t4[lane];
  const float mean = wave_sum32(x.x + x.y + x.z + x.w) * (1.0f / EMB_DIM);
  const float dx = x.x - mean, dy = x.y - mean, dz = x.z - mean, dw = x.w - mean;
  const float var  = wave_sum32(dx * dx + dy * dy + dz * dz + dw * dw) * (1.0f / EMB_DIM);
  const float rstd = rsqrtf(var + LN_EPS);

  const float alpha = (float)(NUM_EMB - 1 - row) * (1.0f / (float)(NUM_EMB - 1));
  const float beta_a = 1.0f - alpha;
  const v4f G = g4[lane], B = b4[lane], A0 = i0[lane], A1 = i1[lane];

  v4f y;
  y.x = dx * rstd * G.x + B.x + alpha * A0.x + beta_a * A1.x;
  y.y = dy * rstd * G.y + B.y + alpha * A0.y + beta_a * A1.y;
  y.z = dz * rstd * G.z + B.z + alpha * A0.z + beta_a * A1.z;
  y.w = dw * rstd * G.w + B.w + alpha * A0.w + beta_a * A1.w;
  t4[lane] = y;
}

// ---------------------------------------------------------------------------
// Kernel 3: gather. One wave32 per output row (512 B), 4 rows per wave for
// memory-level parallelism. Table loads cached (512 KB, L2-resident);
// output stream stored non-temporally to avoid evicting the table.
// ---------------------------------------------------------------------------
__global__ __launch_bounds__(256) void cie_gather_kernel(
    const float* __restrict__ tbl, const long long* __restrict__ idx,
    float* __restrict__ out, int n) {
  const int lane = threadIdx.x & 31;
  const long long wid = (long long)blockIdx.x * (blockDim.x >> 5) + (threadIdx.x >> 5);
  const long long base = wid * 4;
  const v4f* t4 = (const v4f*)tbl;
  v4f* o4 = (v4f*)out;

  if (base + 3 < (long long)n) {
    const long long j0 = idx[base + 0];
    const long long j1 = idx[base + 1];
    const long long j2 = idx[base + 2];
    const long long j3 = idx[base + 3];
    const v4f v0 = t4[j0 * 32 + lane];
    const v4f v1 = t4[j1 * 32 + lane];
    const v4f v2 = t4[j2 * 32 + lane];
    const v4f v3 = t4[j3 * 32 + lane];
    __builtin_nontemporal_store(v0, &o4[(base + 0) * 32 + lane]);
    __builtin_nontemporal_store(v1, &o4[(base + 1) * 32 + lane]);
    __builtin_nontemporal_store(v2, &o4[(base + 2) * 32 + lane]);
    __builtin_nontemporal_store(v3, &o4[(base + 3) * 32 + lane]);
  } else {
    for (long long e = base; e < (long long)n; ++e) {
      const long long j = idx[e];
      __builtin_nontemporal_store(t4[j * 32 + lane], &o4[e * 32 + lane]);
    }
  }
}

// ---------------------------------------------------------------------------
extern "C" void kernel_launch(void* const* d_in, const int* in_sizes, int n_in,
                              void* d_out, int out_size, void* d_ws, size_t ws_size,
                              hipStream_t stream) {
  const float*     E     = (const float*)d_in[0];      // [1024,128]
  const float*     IE    = (const float*)d_in[1];      // [2,128]
  const float*     gamma = (const float*)d_in[2];      // [128]
  const float*     beta  = (const float*)d_in[3];      // [128]
  const long long* idx   = (const long long*)d_in[4];  // [4096*200] int64
  float* out = (float*)d_out;

  float* table = (float*)d_ws;                         // 1024*128*4 = 512 KB
  const int n_entries = in_sizes[4];                   // 819200

  // 1) blocked WMMA prefix-sum -> table (fully overwritten each call)
  cie_cumsum_wmma_kernel<<<1, 256, 0, stream>>>(E, table);

  // 2) LayerNorm + interpolation, in place (1024 rows, 8 rows/block)
  cie_ln_interp_kernel<<<NUM_EMB / 8, 256, 0, stream>>>(table, IE, gamma, beta);

  // 3) gather: 4 rows per wave, 8 waves per block
  const long long waves  = ((long long)n_entries + 3) / 4;
  const int       blocks = (int)((waves + 7) / 8);
  cie_gather_kernel<<<blocks, 256, 0, stream>>>(table, idx, out, n_entries);
}